// PostProcessor__54838142435678
// MI455X (gfx1250) — compile-verified
//
#include <hip/hip_runtime.h>
#include <hip/hip_bf16.h>
#include <math.h>

#define NIMG  2
#define NPROP 2048
#define NCLS  16
#define NFG   15
#define NDET  100
#define IMG_MAX 1023.0f
#define SCORE_THRESH 0.05f
#define NMS_THRESH 0.5f
#define BBOX_CLIP 4.135166556742356f   // log(1000/16)

typedef float v2f __attribute__((ext_vector_type(2)));
typedef float v8f __attribute__((ext_vector_type(8)));

// ---------------------------------------------------------------------------
// Kernel 1: softmax (denominator via V_WMMA_F32_16X16X4_F32) + box decode.
// One wave (32 lanes) handles a tile of 16 proposals. Lane l: proposal
// col = l&15, half = l>>4. Each lane owns 8 classes: {4j + 2*half, +1}.
// WMMA: D = ones(16x4) x E(4x16) accumulated over 4 chunks -> every D VGPR on
// lane l equals the softmax denominator of proposal (l&15).
// ---------------------------------------------------------------------------
__global__ __launch_bounds__(32) void k_decode_softmax(
    const float* __restrict__ logits,   // [B,N,16]
    const float* __restrict__ deltas,   // [B,N,16*5]
    const float* __restrict__ props,    // [B,N,4]
    float* __restrict__ boxesDec,       // [B,15,N,4]
    float* __restrict__ scoresDec,      // [B,15,N]
    float* __restrict__ thetaDec)       // [B,15,N]
{
    const int wave = blockIdx.x;
    const int b    = wave / (NPROP / 16);
    const int tile = wave % (NPROP / 16);
    const int lane = threadIdx.x;
    const int col  = lane & 15;
    const int half = lane >> 4;
    const int n    = tile * 16 + col;

    // Load all 16 logits of this proposal; row max for numerical parity
    const float* lg = logits + ((size_t)b * NPROP + n) * NCLS;
    float l[NCLS];
    float rowmax = -3.4e38f;
#pragma unroll
    for (int k = 0; k < NCLS; ++k) { l[k] = lg[k]; rowmax = fmaxf(rowmax, l[k]); }

    // This lane's 8 exp values (classes 4j + 2*half + {0,1})
    float e[8];
#pragma unroll
    for (int j = 0; j < 4; ++j) {
        int k0 = 4 * j + 2 * half;
        e[2 * j]     = expf(l[k0]     - rowmax);
        e[2 * j + 1] = expf(l[k0 + 1] - rowmax);
    }

    // Denominator via 4 chained f32 WMMAs, A = ones (layout-invariant trick)
    v8f acc = {0.f, 0.f, 0.f, 0.f, 0.f, 0.f, 0.f, 0.f};
    v2f av;  av.x = 1.0f; av.y = 1.0f;
#pragma unroll
    for (int j = 0; j < 4; ++j) {
        v2f bv;  bv.x = e[2 * j]; bv.y = e[2 * j + 1];
        acc = __builtin_amdgcn_wmma_f32_16x16x4_f32(
            /*neg_a=*/false, av, /*neg_b=*/false, bv,
            /*c_mod=*/(short)0, acc, /*reuse_a=*/false, /*reuse_b=*/false);
    }
    const float denom  = acc[0];          // all 8 elements identical per lane
    const float rdenom = 1.0f / denom;

    // Proposal geometry
    const float* pb = props + ((size_t)b * NPROP + n) * 4;
    const float px1 = pb[0], py1 = pb[1], px2 = pb[2], py2 = pb[3];
    const float w  = px2 - px1 + 1.0f;
    const float h  = py2 - py1 + 1.0f;
    const float cx = px1 + 0.5f * w;
    const float cy = py1 + 0.5f * h;

    const float* dl = deltas + ((size_t)b * NPROP + n) * (NCLS * 5);

#pragma unroll
    for (int j = 0; j < 4; ++j) {
#pragma unroll
        for (int t = 0; t < 2; ++t) {
            const int k = 4 * j + 2 * half + t;
            if (k == 0) continue;                      // skip background
            const float prob = e[2 * j + t] * rdenom;
            const float* d = dl + k * 5;
            const float dx = d[0] / 10.0f;
            const float dy = d[1] / 10.0f;
            const float dw = fminf(d[2] / 5.0f, BBOX_CLIP);
            const float dh = fminf(d[3] / 5.0f, BBOX_CLIP);
            const float th = d[4];
            const float pcx = dx * w + cx;
            const float pcy = dy * h + cy;
            const float qw  = expf(dw) * w;
            const float qh  = expf(dh) * h;
            float bx1 = fminf(fmaxf(pcx - 0.5f * qw, 0.f), IMG_MAX);
            float by1 = fminf(fmaxf(pcy - 0.5f * qh, 0.f), IMG_MAX);
            float bx2 = fminf(fmaxf(pcx + 0.5f * qw - 1.0f, 0.f), IMG_MAX);
            float by2 = fminf(fmaxf(pcy + 0.5f * qh - 1.0f, 0.f), IMG_MAX);
            const size_t ci = (size_t)(b * NFG + (k - 1)) * NPROP + n;
            scoresDec[ci] = prob;
            thetaDec[ci]  = th;
            boxesDec[ci * 4 + 0] = bx1;
            boxesDec[ci * 4 + 1] = by1;
            boxesDec[ci * 4 + 2] = bx2;
            boxesDec[ci * 4 + 3] = by2;
        }
    }
}

// ---------------------------------------------------------------------------
// Kernel 2: per-(image,class) descending bitonic argsort of 2048 scores.
// Comparator: key desc, index asc (stable w.r.t. argsort(-scores)).
// ---------------------------------------------------------------------------
__global__ __launch_bounds__(1024) void k_sort(const float* __restrict__ scoresDec,
                                               int* __restrict__ perm)
{
    __shared__ float key[NPROP];
    __shared__ int   sidx[NPROP];
    const int    bc   = blockIdx.x;              // b*15 + c
    const size_t base = (size_t)bc * NPROP;

    for (int t = threadIdx.x; t < NPROP; t += blockDim.x) {
        key[t]  = scoresDec[base + t];
        sidx[t] = t;
    }
    __syncthreads();

    for (int k = 2; k <= NPROP; k <<= 1) {
        for (int j = k >> 1; j > 0; j >>= 1) {
            for (int t = threadIdx.x; t < NPROP; t += blockDim.x) {
                const int ixj = t ^ j;
                if (ixj > t) {
                    const float ka = key[t], kb = key[ixj];
                    const int   ia = sidx[t], ib = sidx[ixj];
                    const bool aFirst = (ka > kb) || (ka == kb && ia < ib);
                    const bool desc   = ((t & k) == 0);
                    if (desc ? !aFirst : aFirst) {
                        key[t] = kb; key[ixj] = ka;
                        sidx[t] = ib; sidx[ixj] = ia;
                    }
                }
            }
            __syncthreads();
        }
    }
    for (int t = threadIdx.x; t < NPROP; t += blockDim.x) perm[base + t] = sidx[t];
}

// ---------------------------------------------------------------------------
// Kernel 3: greedy NMS per (image,class). Sorted tile cached in LDS, serial
// scan over i with parallel suppression of j>i. Writes score-or-(-1) per slot.
// ---------------------------------------------------------------------------
__global__ __launch_bounds__(256) void k_nms(const float* __restrict__ boxesDec,
                                             const float* __restrict__ scoresDec,
                                             const int*   __restrict__ perm,
                                             float*       __restrict__ keepScore)
{
    __shared__ float sx1[NPROP], sy1[NPROP], sx2[NPROP], sy2[NPROP];
    __shared__ float sar[NPROP], ssc[NPROP];
    __shared__ unsigned char supp[NPROP];
    __shared__ int s_cond;

    const int    bc   = blockIdx.x;
    const size_t base = (size_t)bc * NPROP;

    for (int t = threadIdx.x; t < NPROP; t += blockDim.x) {
        const int p = perm[base + t];
        const float* bx = boxesDec + (base + p) * 4;
        const float x1 = bx[0], y1 = bx[1], x2 = bx[2], y2 = bx[3];
        sx1[t] = x1; sy1[t] = y1; sx2[t] = x2; sy2[t] = y2;
        sar[t] = (x2 - x1 + 1.0f) * (y2 - y1 + 1.0f);
        ssc[t] = scoresDec[base + p];
        supp[t] = 0;
    }
    __syncthreads();

    for (int i = 0; i < NPROP; ++i) {
        if (threadIdx.x == 0) {
            const int cond = (ssc[i] > SCORE_THRESH) && (supp[i] == 0);
            s_cond = cond;
            keepScore[base + i] = cond ? ssc[i] : -1.0f;
        }
        __syncthreads();
        if (s_cond) {
            const float ix1 = sx1[i], iy1 = sy1[i], ix2 = sx2[i], iy2 = sy2[i];
            const float iar = sar[i];
            for (int t = i + 1 + (int)threadIdx.x; t < NPROP; t += blockDim.x) {
                if (!supp[t]) {
                    float ww = fminf(ix2, sx2[t]) - fmaxf(ix1, sx1[t]) + 1.0f;
                    float hh = fminf(iy2, sy2[t]) - fmaxf(iy1, sy1[t]) + 1.0f;
                    ww = fmaxf(ww, 0.0f); hh = fmaxf(hh, 0.0f);
                    const float inter = ww * hh;
                    const float iou   = inter / (iar + sar[t] - inter);
                    if (iou > NMS_THRESH) supp[t] = 1;
                }
            }
        }
        __syncthreads();
    }
}

// ---------------------------------------------------------------------------
// Kernel 4: per-image top-100 by iterated parallel argmax (ties -> low index),
// then gather boxes/theta/labels and write the flat-float output tuple.
// ---------------------------------------------------------------------------
__global__ __launch_bounds__(1024) void k_topk(const float* __restrict__ boxesDec,
                                               const float* __restrict__ thetaDec,
                                               const int*   __restrict__ perm,
                                               float*       __restrict__ keepScore,
                                               float*       __restrict__ out)
{
    __shared__ float rk[1024];
    __shared__ int   ri[1024];
    const int b = blockIdx.x;
    const int FLATN = NFG * NPROP;                 // 30720
    float* ks = keepScore + (size_t)b * FLATN;

    const int BOX_OFF = 0;
    const int SC_OFF  = NIMG * NDET * 4;           // 800
    const int TH_OFF  = SC_OFF + NIMG * NDET;      // 1000
    const int LB_OFF  = TH_OFF + NIMG * NDET;      // 1200
    const int VA_OFF  = LB_OFF + NIMG * NDET;      // 1400

    for (int d = 0; d < NDET; ++d) {
        float best = -1e30f;
        int   bi   = 0x7fffffff;
        for (int t = threadIdx.x; t < FLATN; t += blockDim.x) {
            const float v = ks[t];
            if (v > best || (v == best && t < bi)) { best = v; bi = t; }
        }
        rk[threadIdx.x] = best; ri[threadIdx.x] = bi;
        __syncthreads();
        for (int s = blockDim.x >> 1; s > 0; s >>= 1) {
            if ((int)threadIdx.x < s) {
                const float vo = rk[threadIdx.x + s];
                const int   io = ri[threadIdx.x + s];
                if (vo > rk[threadIdx.x] ||
                    (vo == rk[threadIdx.x] && io < ri[threadIdx.x])) {
                    rk[threadIdx.x] = vo; ri[threadIdx.x] = io;
                }
            }
            __syncthreads();
        }
        if (threadIdx.x == 0) {
            const float s0 = rk[0];
            const int   i0 = ri[0];
            ks[i0] = -2.0f;                         // remove from further picks
            const bool valid = s0 > 0.0f;
            const int c15 = i0 / NPROP;
            const int pos = i0 % NPROP;
            const size_t cb = (size_t)(b * NFG + c15) * NPROP;
            const int p = perm[cb + pos];
            const float* bx = boxesDec + (cb + p) * 4;
            const int o = b * NDET + d;
            out[BOX_OFF + (size_t)o * 4 + 0] = valid ? bx[0] : 0.f;
            out[BOX_OFF + (size_t)o * 4 + 1] = valid ? bx[1] : 0.f;
            out[BOX_OFF + (size_t)o * 4 + 2] = valid ? bx[2] : 0.f;
            out[BOX_OFF + (size_t)o * 4 + 3] = valid ? bx[3] : 0.f;
            out[SC_OFF + o] = valid ? s0 : 0.f;
            out[TH_OFF + o] = valid ? thetaDec[cb + p] : 0.f;
            out[LB_OFF + o] = valid ? (float)(c15 + 1) : 0.f;
            out[VA_OFF + o] = valid ? 1.f : 0.f;
        }
        __syncthreads();
    }
}

extern "C" void kernel_launch(void* const* d_in, const int* in_sizes, int n_in,
                              void* d_out, int out_size, void* d_ws, size_t ws_size,
                              hipStream_t stream) {
    const float* logits = (const float*)d_in[0];   // [B,N,16]
    const float* deltas = (const float*)d_in[1];   // [B,N,80]
    const float* props  = (const float*)d_in[2];   // [B,N,4]

    float* ws        = (float*)d_ws;
    float* boxesDec  = ws;                                          // B*15*N*4
    float* scoresDec = boxesDec + (size_t)NIMG * NFG * NPROP * 4;   // B*15*N
    float* thetaDec  = scoresDec + (size_t)NIMG * NFG * NPROP;      // B*15*N
    int*   perm      = (int*)(thetaDec + (size_t)NIMG * NFG * NPROP);
    float* keepScore = (float*)(perm + (size_t)NIMG * NFG * NPROP); // B*15*N

    k_decode_softmax<<<NIMG * (NPROP / 16), 32, 0, stream>>>(
        logits, deltas, props, boxesDec, scoresDec, thetaDec);
    k_sort<<<NIMG * NFG, 1024, 0, stream>>>(scoresDec, perm);
    k_nms<<<NIMG * NFG, 256, 0, stream>>>(boxesDec, scoresDec, perm, keepScore);
    k_topk<<<NIMG, 1024, 0, stream>>>(boxesDec, thetaDec, perm, keepScore,
                                      (float*)d_out);
}